// smartFNN_35914516529312
// MI455X (gfx1250) — compile-verified
//
#include <hip/hip_runtime.h>
#include <hip/hip_bf16.h>

// ---- problem constants (match reference) ----
constexpr int kB   = 2;
constexpr int kN   = 256;
constexpr int kD   = 256;
constexpr int kGL  = 1024;
constexpr int kFC1 = 1024;
constexpr int kFC2 = 512;
constexpr int kT1  = 512;
constexpr int kT2  = 128;

typedef __attribute__((ext_vector_type(2))) float v2f;
typedef __attribute__((ext_vector_type(8))) float v8f;

#if defined(__has_builtin)
#if __has_builtin(__builtin_amdgcn_tensor_load_to_lds) && \
    __has_builtin(__builtin_amdgcn_s_wait_tensorcnt)
#define HAVE_TDM 1
#endif
#endif

__device__ __forceinline__ v8f wmma4(v2f a, v2f b, v8f c) {
  // D = A(16x4, f32) * B(4x16, f32) + C(16x16, f32)
  return __builtin_amdgcn_wmma_f32_16x16x4_f32(
      /*neg_a=*/false, a, /*neg_b=*/false, b,
      /*c_mod=*/(short)0, c, /*reuse_a=*/false, /*reuse_b=*/false);
}

__device__ __forceinline__ float lrelu(float x) {
  return fmaxf(x, 0.01f * x);   // leaky_relu(x, 0.01) for all x
}

#ifdef HAVE_TDM
// 1D contiguous TDM copy: global -> LDS, nelem fp32 elements (nelem <= 65535).
// D# group0: count=1 | lds_addr | global_addr | type=2 ("image")
// D# group1: data_size=4B, tensor_dim0 = tile_dim0 = nelem, dim1 = 1
// 6-arg builtin form (clang-23 / therock headers): (g0, g1, g2, g3, g4, cpol)
__device__ __forceinline__ void tdm_load_1d(unsigned lds_addr, const void* gptr,
                                            int nelem) {
  typedef unsigned int u32x4 __attribute__((ext_vector_type(4)));
  typedef int i32x8 __attribute__((ext_vector_type(8)));
  typedef int i32x4 __attribute__((ext_vector_type(4)));
  const unsigned long long ga = (unsigned long long)(uintptr_t)gptr;
  u32x4 g0;
  g0.x = 1u;                                             // count=1, user mode
  g0.y = lds_addr;                                       // bits[63:32]
  g0.z = (unsigned)ga;                                   // global_addr[31:0]
  g0.w = (unsigned)((ga >> 32) & 0x01FFFFFFu) | (2u << 30);  // addr[56:32]|type=2
  i32x8 g1;
  g1[0] = (2 << 16);                                     // data_size = 4 bytes
  g1[1] = (nelem & 0xFFFF) << 16;                        // tensor_dim0[15:0]
  g1[2] = ((unsigned)nelem >> 16) | (1 << 16);           // tensor_dim0[31:16] | tensor_dim1=1
  g1[3] = (nelem & 0xFFFF) << 16;                        // tile_dim0
  g1[4] = 1;                                             // tile_dim1 = 1
  g1[5] = nelem;                                         // tensor_dim0_stride[31:0]
  g1[6] = 0;
  g1[7] = 0;
  const i32x4 z4 = {0, 0, 0, 0};
  const i32x8 z8 = {0, 0, 0, 0, 0, 0, 0, 0};
  __builtin_amdgcn_tensor_load_to_lds(g0, g1, z4, z4, z8, 0);
}
#endif

// ---------------------------------------------------------------------------
// K1: g1[b, gl] = sum_n lrelu( (in[b] @ Wg)[n, gl] + bg[gl] )
// ---------------------------------------------------------------------------
__global__ void k_gsum(const float* __restrict__ in, const float* __restrict__ Wg,
                       const float* __restrict__ bg, float* __restrict__ g1) {
  const int wid  = blockIdx.x;
  const int b    = wid / (kGL / 16);
  const int gl0  = (wid % (kGL / 16)) * 16;
  const int lane = threadIdx.x;
  const int col  = lane & 15;
  const int kh   = lane >> 4;

  const float* inb = in + (size_t)b * kN * kD;
  const float bias = bg[gl0 + col];

  float colsum = 0.f;
  for (int n0 = 0; n0 < kN; n0 += 16) {
    v8f c = {};
    const float* arow = inb + (size_t)(n0 + col) * kD;
    for (int k0 = 0; k0 < kD; k0 += 4) {
      const int ka = k0 + 2 * kh;
      v2f a = *(const v2f*)(arow + ka);                 // A: (M=col, K=ka..ka+1)
      v2f bm;
      bm.x = Wg[(size_t)ka * kGL + gl0 + col];          // B: (K=ka,   N=col)
      bm.y = Wg[(size_t)(ka + 1) * kGL + gl0 + col];    // B: (K=ka+1, N=col)
      c = wmma4(a, bm, c);
    }
#pragma unroll
    for (int v = 0; v < 8; ++v) colsum += lrelu(c[v] + bias);
  }
  colsum += __shfl_xor(colsum, 16, 32);
  if (lane < 16) g1[(size_t)b * kGL + gl0 + lane] = colsum;
}

// ---------------------------------------------------------------------------
// K2: tiny dense: out[b,o] = act?( x[b]@W + bias ), M = kB only
// ---------------------------------------------------------------------------
__global__ void k_fc(const float* __restrict__ x, const float* __restrict__ W,
                     const float* __restrict__ bias, float* __restrict__ out,
                     int K, int O, int doAct) {
  const int idx = blockIdx.x * blockDim.x + threadIdx.x;
  if (idx >= kB * O) return;
  const int b = idx / O, o = idx % O;
  float acc = bias ? bias[o] : 0.f;
  const float* xb = x + (size_t)b * K;
  for (int k = 0; k < K; ++k) acc = fmaf(xb[k], W[(size_t)k * O + o], acc);
  out[idx] = doAct ? lrelu(acc) : acc;
}

// ---------------------------------------------------------------------------
// K3: pj = in @ Wt1[0:D], pi = in @ Wt1[D:2D]
// ---------------------------------------------------------------------------
__global__ void k_proj(const float* __restrict__ in, const float* __restrict__ Wt1,
                       float* __restrict__ pj, float* __restrict__ pi) {
  const int wid   = blockIdx.x;
  const int ttile = wid & 31;
  const int ntile = (wid >> 5) & 15;
  const int b     = (wid >> 9) & 1;
  const int which = wid >> 10;
  const int lane  = threadIdx.x;
  const int col   = lane & 15;
  const int kh    = lane >> 4;

  const int n0 = ntile * 16, t0 = ttile * 16;
  const float* Wm   = Wt1 + (size_t)which * kD * kT1;
  const float* arow = in + ((size_t)(b * kN + n0 + col)) * kD;

  v8f c = {};
  for (int k0 = 0; k0 < kD; k0 += 4) {
    const int ka = k0 + 2 * kh;
    v2f a = *(const v2f*)(arow + ka);
    v2f bm;
    bm.x = Wm[(size_t)ka * kT1 + t0 + col];
    bm.y = Wm[(size_t)(ka + 1) * kT1 + t0 + col];
    c = wmma4(a, bm, c);
  }
  float* dst = (which ? pi : pj) + (size_t)b * kN * kT1;
#pragma unroll
  for (int v = 0; v < 8; ++v) {
    const int m = v + 8 * kh;
    dst[(size_t)(n0 + m) * kT1 + t0 + col] = c[v];
  }
}

// ---------------------------------------------------------------------------
// K3b: pack Wt2 into K-pair interleaved layout (done once, reused by all 512
// k_pair blocks): w2p[((k/2)*kT2 + n)*2 + (k&1)] = Wt2[k*kT2 + n]
// Each KCH-chunk of packed data is contiguous -> TDM 1D tile copy.
// ---------------------------------------------------------------------------
__global__ void k_packw2(const float* __restrict__ Wt2, float* __restrict__ w2p) {
  const int idx = blockIdx.x * blockDim.x + threadIdx.x;   // over kT1*kT2
  const int k = idx >> 7, n = idx & (kT2 - 1);
  w2p[(((k >> 1) * kT2 + n) << 1) | (k & 1)] = Wt2[idx];
}

// ---------------------------------------------------------------------------
// K4 (hot): per workgroup = one (b, i). M = j (256), K = kT1 (512), N = kT2.
//   A[j,k] = lrelu(pj[b,j,k] + c[k]),  c = pi[b,i,:] + pg[b,:] + bt1 (LDS)
//   C = A @ Wt2 ;  out[b,i,j] = lrelu(C + bt2) @ Wt3 + bt3  (fused epilogue)
// Wt2 chunks (packed) streamed into double-buffered LDS by the TDM (wave 0
// issues tensor_load_to_lds; s_wait_tensorcnt + barrier pipelines it against
// the WMMA stream of all 8 waves). Fallback: synchronous cooperative staging.
// ---------------------------------------------------------------------------
constexpr int KCH = 32;
constexpr int NCH = kT1 / KCH;              // 16 chunks
constexpr int CHUNK_ELEMS = KCH * kT2;      // 4096 floats = 16 KB

__global__ __launch_bounds__(256) void k_pair(
    const float* __restrict__ pj, const float* __restrict__ pi,
    const float* __restrict__ pg, const float* __restrict__ bt1,
    const float* __restrict__ w2p, const float* __restrict__ bt2,
    const float* __restrict__ Wt3, const float* __restrict__ bt3,
    float* __restrict__ out) {
  __shared__ float cvec[kT1];                 // 2 KB
  __shared__ float Bs[2][CHUNK_ELEMS];        // 2 x 16 KB, double buffered

  const int bi  = blockIdx.x;                 // b*kN + i
  const int b   = bi / kN;
  const int tid = threadIdx.x;
  const int wave = tid >> 5, lane = tid & 31;
  const int col = lane & 15, kh = lane >> 4;

#ifdef HAVE_TDM
  if (wave == 0) tdm_load_1d((unsigned)(uintptr_t)&Bs[0][0], w2p, CHUNK_ELEMS);
#endif

  // stage the per-(b,i) additive K-vector
  for (int k = tid; k < kT1; k += 256)
    cvec[k] = pi[(size_t)bi * kT1 + k] + pg[(size_t)b * kT1 + k] + bt1[k];

  v8f acc[2][8];
#pragma unroll
  for (int mt = 0; mt < 2; ++mt)
#pragma unroll
    for (int nt = 0; nt < 8; ++nt) acc[mt][nt] = (v8f){};

  const float* pjb = pj + (size_t)b * kN * kT1;
  const float* prow0 = pjb + (size_t)(wave * 32 + col) * kT1;        // mt=0
  const float* prow1 = pjb + (size_t)(wave * 32 + 16 + col) * kT1;   // mt=1

  for (int c = 0; c < NCH; ++c) {
    const int kc = c * KCH;
#ifdef HAVE_TDM
    if (wave == 0) {
      if (c + 1 < NCH) {
        // prefetch next chunk into the other buffer, then retire current one
        tdm_load_1d((unsigned)(uintptr_t)&Bs[(c + 1) & 1][0],
                    w2p + (size_t)(c + 1) * CHUNK_ELEMS, CHUNK_ELEMS);
        __builtin_amdgcn_s_wait_tensorcnt(1);   // in-order: chunk c complete
      } else {
        __builtin_amdgcn_s_wait_tensorcnt(0);
      }
    }
    __syncthreads();                            // publish Bs[c&1] (+ cvec at c=0)
    const float* Bc = &Bs[c & 1][0];
#else
    __syncthreads();                            // protect Bs reuse (+ cvec at c=0)
    for (int idx = tid; idx < CHUNK_ELEMS; idx += 256) {
      const int kk = idx >> 7, n = idx & (kT2 - 1);
      Bs[0][(((kk >> 1) * kT2 + n) << 1) | (kk & 1)] =
          w2p[(size_t)kc * kT2 + idx];          // packed source, scattered again by the pack index
    }
    __syncthreads();
    const float* Bc = &Bs[0][0];
#endif

    for (int ks = 0; ks < KCH; ks += 4) {
      const int kl = ks + 2 * kh;               // even: this lane's K-pair base
      v2f bmv[8];                               // hoisted across both M-tiles
#pragma unroll
      for (int nt = 0; nt < 8; ++nt)
        bmv[nt] = *(const v2f*)(Bc + (((kl >> 1) * kT2 + nt * 16 + col) << 1));

      const int ka = kc + kl;
      const v2f cv = *(const v2f*)(cvec + ka);
      {
        const v2f p = *(const v2f*)(prow0 + ka);
        v2f a; a.x = lrelu(p.x + cv.x); a.y = lrelu(p.y + cv.y);
#pragma unroll
        for (int nt = 0; nt < 8; ++nt) acc[0][nt] = wmma4(a, bmv[nt], acc[0][nt]);
      }
      {
        const v2f p = *(const v2f*)(prow1 + ka);
        v2f a; a.x = lrelu(p.x + cv.x); a.y = lrelu(p.y + cv.y);
#pragma unroll
        for (int nt = 0; nt < 8; ++nt) acc[1][nt] = wmma4(a, bmv[nt], acc[1][nt]);
      }
    }
#ifdef HAVE_TDM
    __syncthreads();   // all waves done with Bs[c&1] before it is overwritten
#endif
  }

  // fused epilogue: y = lrelu(C + bt2); out_row = y @ Wt3 + bt3
  const float bt3v = bt3[0];
#pragma unroll
  for (int mt = 0; mt < 2; ++mt) {
    float rowsum[8];
#pragma unroll
    for (int v = 0; v < 8; ++v) rowsum[v] = 0.f;
#pragma unroll
    for (int nt = 0; nt < 8; ++nt) {
      const int n  = nt * 16 + col;
      const float b2 = bt2[n], w3 = Wt3[n];
#pragma unroll
      for (int v = 0; v < 8; ++v) rowsum[v] += lrelu(acc[mt][nt][v] + b2) * w3;
    }
#pragma unroll
    for (int v = 0; v < 8; ++v) {
      rowsum[v] += __shfl_xor(rowsum[v], 1, 32);
      rowsum[v] += __shfl_xor(rowsum[v], 2, 32);
      rowsum[v] += __shfl_xor(rowsum[v], 4, 32);
      rowsum[v] += __shfl_xor(rowsum[v], 8, 32);
    }
    if (col == 0) {
      const int j0 = (wave * 2 + mt) * 16 + kh * 8;
#pragma unroll
      for (int v = 0; v < 8; ++v)
        out[(size_t)bi * kN + j0 + v] = rowsum[v] + bt3v;
    }
  }
}

// ---------------------------------------------------------------------------
extern "C" void kernel_launch(void* const* d_in, const int* in_sizes, int n_in,
                              void* d_out, int out_size, void* d_ws, size_t ws_size,
                              hipStream_t stream) {
  (void)in_sizes; (void)n_in; (void)out_size; (void)ws_size;
  const float* in_t = (const float*)d_in[0];
  const float* Wg   = (const float*)d_in[1];
  const float* bg   = (const float*)d_in[2];
  const float* Wf1  = (const float*)d_in[3];
  const float* bf1  = (const float*)d_in[4];
  const float* Wf2  = (const float*)d_in[5];
  const float* bf2  = (const float*)d_in[6];
  const float* Wt1  = (const float*)d_in[7];
  const float* bt1  = (const float*)d_in[8];
  const float* Wt2  = (const float*)d_in[9];
  const float* bt2  = (const float*)d_in[10];
  const float* Wt3  = (const float*)d_in[11];
  const float* bt3  = (const float*)d_in[12];
  float* out = (float*)d_out;

  float* ws  = (float*)d_ws;
  float* g1  = ws;                       // kB*kGL    = 2048
  float* g2  = g1 + kB * kGL;            // kB*kFC1   = 2048
  float* g3  = g2 + kB * kFC1;           // kB*kFC2   = 1024
  float* pg  = g3 + kB * kFC2;           // kB*kT1    = 1024
  float* pj  = pg + kB * kT1;            // kB*kN*kT1 = 262144
  float* pi  = pj + kB * kN * kT1;       // kB*kN*kT1 = 262144
  float* w2p = pi + kB * kN * kT1;       // kT1*kT2   = 65536

  // 1) g1 = sum_n lrelu(in@Wg + bg)
  k_gsum<<<kB * kGL / 16, 32, 0, stream>>>(in_t, Wg, bg, g1);
  // 2) g2 = lrelu(g1@Wf1 + bf1) ; g3 = lrelu(g2@Wf2 + bf2)
  k_fc<<<(kB * kFC1 + 255) / 256, 256, 0, stream>>>(g1, Wf1, bf1, g2, kGL, kFC1, 1);
  k_fc<<<(kB * kFC2 + 255) / 256, 256, 0, stream>>>(g2, Wf2, bf2, g3, kFC1, kFC2, 1);
  // 3) pg = g3 @ Wgm
  k_fc<<<(kB * kT1 + 255) / 256, 256, 0, stream>>>(g3, Wt1 + (size_t)2 * kD * kT1,
                                                   nullptr, pg, kFC2, kT1, 0);
  // 4) pj = in @ Wj ; pi = in @ Wi ; pack Wt2
  k_proj<<<2 * kB * (kN / 16) * (kT1 / 16), 32, 0, stream>>>(in_t, Wt1, pj, pi);
  k_packw2<<<kT1 * kT2 / 256, 256, 0, stream>>>(Wt2, w2p);
  // 5) fused pair stage -> out[b,i,j]
  k_pair<<<kB * kN, 256, 0, stream>>>(pj, pi, pg, bt1, w2p, bt2, Wt3, bt3, out);
}